// HeteroMessagePassingLayer_26422638805022
// MI455X (gfx1250) — compile-verified
//
#include <hip/hip_runtime.h>
#include <math.h>

typedef float v2f __attribute__((ext_vector_type(2)));
typedef float v8f __attribute__((ext_vector_type(8)));

#define D 128
#define XS_STRIDE 132   // padded LDS row stride (floats): 132 mod 64 banks = 4 -> conflict-free

__device__ __forceinline__ float gelu_exact(float x) {
    return 0.5f * x * (1.0f + erff(x * 0.70710678118654752f));
}

// ---------------------------------------------------------------- zero scratch
__global__ void zero_kernel(float* __restrict__ p, size_t n) {
    size_t i = (size_t)blockIdx.x * blockDim.x + threadIdx.x;
    size_t stride = (size_t)gridDim.x * blockDim.x;
    size_t n4 = n >> 2;
    float4 z = {0.f, 0.f, 0.f, 0.f};
    for (size_t j = i; j < n4; j += stride) ((float4*)p)[j] = z;
    for (size_t j = (n4 << 2) + i; j < n; j += stride) p[j] = 0.0f;
}

// ---------------------------------------------------------------- T = X @ W^T
// X: [n, 128] row-major, W: [128, 128] torch layout [out, in].
// Block = 256 threads (8 waves). Block computes a 16-row x 128-col tile of T.
// Wave w computes columns [16w, 16w+16) with a K-loop of V_WMMA_F32_16X16X4_F32.
__global__ void gemm_xwt_kernel(const float* __restrict__ X,
                                const float* __restrict__ W,
                                float* __restrict__ T, int n) {
    __shared__ __align__(16) float xs[16 * XS_STRIDE];
    const int row0 = blockIdx.x * 16;
    const int t = threadIdx.x;
    const bool full = (row0 + 16 <= n);

    // stage 16x128 X tile into LDS (coalesced)
    if (full) {
        for (int i = t; i < 16 * D; i += 256) {
            int m = i >> 7, k = i & 127;
            xs[m * XS_STRIDE + k] = X[(size_t)(row0 + m) * D + k];
        }
    } else {
        for (int i = t; i < 16 * D; i += 256) {
            int m = i >> 7, k = i & 127;
            int r = row0 + m; if (r > n - 1) r = n - 1;
            xs[m * XS_STRIDE + k] = X[(size_t)r * D + k];
        }
    }
    __syncthreads();

    const int wave = t >> 5, lane = t & 31;
    const int g = lane >> 4;        // K-half selector (lanes 0-15 vs 16-31)
    const int mn = lane & 15;       // M row (for A) / N col (for B)
    const int ncol0 = wave * 16;

    v8f acc = {};
    #pragma unroll 8
    for (int k0 = 0; k0 < D; k0 += 4) {
        // A fragment: row mn, K = k0+2g, k0+2g+1  (8B-aligned LDS read)
        v2f a = *(const v2f*)&xs[mn * XS_STRIDE + k0 + 2 * g];
        // B fragment: B[k][n] = W[n][k] -> contiguous row of W (hits WGP$/L2)
        v2f b = *(const v2f*)&W[(size_t)(ncol0 + mn) * D + k0 + 2 * g];
        acc = __builtin_amdgcn_wmma_f32_16x16x4_f32(false, a, false, b,
                                                    (short)0, acc, false, false);
    }

    // C layout: VGPR r -> M = r + 8*g, N = ncol0 + mn
    float* tp = T + (size_t)(row0 + 8 * g) * D + ncol0 + mn;
    if (full) {
        #pragma unroll
        for (int r = 0; r < 8; r++) tp[(size_t)r * D] = acc[r];
    } else {
        #pragma unroll
        for (int r = 0; r < 8; r++)
            if (row0 + r + 8 * g < n) tp[(size_t)r * D] = acc[r];
    }
}

// ---------------------------------------------------------------- edge scatter
// One wave per edge: lanes read a coalesced 512B row of T[src], atomic-add
// into agg[tgt]; lane 0 bumps the degree counter.
__global__ void scatter_kernel(const float* __restrict__ T,
                               const int* __restrict__ ei, int E,
                               float* __restrict__ agg,
                               float* __restrict__ deg) {
    int e = blockIdx.x * 8 + (threadIdx.x >> 5);
    if (e >= E) return;
    int lane = threadIdx.x & 31;
    int src = ei[e];
    int tgt = ei[E + e];
    const float4 v = *(const float4*)(T + (size_t)src * D + lane * 4);
    float* dst = agg + (size_t)tgt * D + lane * 4;
    atomicAdd(dst + 0, v.x);
    atomicAdd(dst + 1, v.y);
    atomicAdd(dst + 2, v.z);
    atomicAdd(dst + 3, v.w);
    if (lane == 0) atomicAdd(deg + tgt, 1.0f);
}

// -------------------------------------------- self-linear + msg + LN + GELU
// Block = 128 threads (4 waves), 16 rows per block.
// Wave w computes two 16x16 WMMA tiles (cols 32w..32w+31) of X @ Wself^T and
// spills the raw result to LDS. The LN stage then does everything else with
// fully-coalesced float4 accesses: x = self + bias + agg/deg, LayerNorm, GELU.
__global__ void self_ln_gelu_kernel(const float* __restrict__ X,
                                    const float* __restrict__ Wself,
                                    const float* __restrict__ bias,
                                    const float* __restrict__ agg,
                                    const float* __restrict__ deg,
                                    const float* __restrict__ gamma,
                                    const float* __restrict__ beta,
                                    float* __restrict__ out, int n) {
    __shared__ __align__(16) float xs[16 * XS_STRIDE];
    __shared__ __align__(16) float ys[16 * XS_STRIDE];
    const int row0 = blockIdx.x * 16;
    const int t = threadIdx.x;
    const bool full = (row0 + 16 <= n);

    if (full) {
        for (int i = t; i < 16 * D; i += 128) {
            int m = i >> 7, k = i & 127;
            xs[m * XS_STRIDE + k] = X[(size_t)(row0 + m) * D + k];
        }
    } else {
        for (int i = t; i < 16 * D; i += 128) {
            int m = i >> 7, k = i & 127;
            int r = row0 + m; if (r > n - 1) r = n - 1;
            xs[m * XS_STRIDE + k] = X[(size_t)r * D + k];
        }
    }
    __syncthreads();

    const int wave = t >> 5, lane = t & 31;
    const int g = lane >> 4, mn = lane & 15;
    const int nc0 = wave * 32, nc1 = wave * 32 + 16;

    v8f acc0 = {}, acc1 = {};
    #pragma unroll 4
    for (int k0 = 0; k0 < D; k0 += 4) {
        v2f a  = *(const v2f*)&xs[mn * XS_STRIDE + k0 + 2 * g];
        v2f b0 = *(const v2f*)&Wself[(size_t)(nc0 + mn) * D + k0 + 2 * g];
        v2f b1 = *(const v2f*)&Wself[(size_t)(nc1 + mn) * D + k0 + 2 * g];
        acc0 = __builtin_amdgcn_wmma_f32_16x16x4_f32(false, a, false, b0,
                                                     (short)0, acc0, false, false);
        acc1 = __builtin_amdgcn_wmma_f32_16x16x4_f32(false, a, false, b1,
                                                     (short)0, acc1, false, false);
    }

    // spill raw self-linear result to LDS (C layout: VGPR r -> M = r + 8g)
    #pragma unroll
    for (int r = 0; r < 8; r++) {
        int m = r + 8 * g;
        ys[m * XS_STRIDE + nc0 + mn] = acc0[r];
        ys[m * XS_STRIDE + nc1 + mn] = acc1[r];
    }
    __syncthreads();

    // bias + mean message + LayerNorm + GELU
    // wave handles rows 4*wave .. 4*wave+3; lane covers 4 consecutive cols
    const float4 gm = *(const float4*)&gamma[lane * 4];
    const float4 bt = *(const float4*)&beta[lane * 4];
    const float4 bs = *(const float4*)&bias[lane * 4];
    for (int rr = wave * 4; rr < wave * 4 + 4; rr++) {
        int row = row0 + rr;
        int rowc = row; if (rowc > n - 1) rowc = n - 1;
        float  rd = 1.0f / fmaxf(deg[rowc], 1.0f);
        float4 av = *(const float4*)&agg[(size_t)rowc * D + lane * 4];
        float4 sv = *(const float4*)&ys[rr * XS_STRIDE + lane * 4];
        float4 x;
        x.x = sv.x + bs.x + av.x * rd;
        x.y = sv.y + bs.y + av.y * rd;
        x.z = sv.z + bs.z + av.z * rd;
        x.w = sv.w + bs.w + av.w * rd;
        float s  = x.x + x.y + x.z + x.w;
        float s2 = x.x * x.x + x.y * x.y + x.z * x.z + x.w * x.w;
        #pragma unroll
        for (int off = 16; off >= 1; off >>= 1) {
            s  += __shfl_xor(s,  off, 32);
            s2 += __shfl_xor(s2, off, 32);
        }
        float mu  = s * (1.0f / D);
        float var = s2 * (1.0f / D) - mu * mu;
        float rs  = rsqrtf(var + 1e-5f);
        float4 o;
        o.x = gelu_exact((x.x - mu) * rs * gm.x + bt.x);
        o.y = gelu_exact((x.y - mu) * rs * gm.y + bt.y);
        o.z = gelu_exact((x.z - mu) * rs * gm.z + bt.z);
        o.w = gelu_exact((x.w - mu) * rs * gm.w + bt.w);
        if (row < n) *(float4*)&out[(size_t)row * D + lane * 4] = o;
    }
}

// ---------------------------------------------------------------- launcher
extern "C" void kernel_launch(void* const* d_in, const int* in_sizes, int n_in,
                              void* d_out, int out_size, void* d_ws, size_t ws_size,
                              hipStream_t stream) {
    const float* emb_cell   = (const float*)d_in[0];
    const float* emb_vertex = (const float*)d_in[1];
    const float* W_cv       = (const float*)d_in[2];
    const float* W_vc       = (const float*)d_in[3];
    const float* W_cc       = (const float*)d_in[4];
    const float* W_self_c   = (const float*)d_in[5];
    const float* b_self_c   = (const float*)d_in[6];
    const float* W_self_v   = (const float*)d_in[7];
    const float* b_self_v   = (const float*)d_in[8];
    const float* g_c        = (const float*)d_in[9];
    const float* beta_c     = (const float*)d_in[10];
    const float* g_v        = (const float*)d_in[11];
    const float* beta_v     = (const float*)d_in[12];
    const int*   ei_cv      = (const int*)d_in[13];
    const int*   ei_vc      = (const int*)d_in[14];
    const int*   ei_cc      = (const int*)d_in[15];

    const int NC = in_sizes[0] / D;
    const int NV = in_sizes[1] / D;
    const int E_cv = in_sizes[13] / 2;
    const int E_vc = in_sizes[14] / 2;
    const int E_cc = in_sizes[15] / 2;

    const size_t nDc = (size_t)NC * D;
    const size_t nDv = (size_t)NV * D;
    const size_t tsz = nDc > nDv ? nDc : nDv;

    float* ws    = (float*)d_ws;
    float* Tbuf  = ws;                 // reused serially for the 3 message GEMMs
    float* agg_c = ws + tsz;
    float* agg_v = agg_c + nDc;
    float* deg_c = agg_v + nDv;
    float* deg_v = deg_c + NC;

    // zero agg_c | agg_v | deg_c | deg_v (contiguous)
    const size_t zn = nDc + nDv + (size_t)NC + (size_t)NV;
    zero_kernel<<<2048, 256, 0, stream>>>(agg_c, zn);

    const int rtC = (NC + 15) / 16;
    const int rtV = (NV + 15) / 16;

    // cell -> vertex messages
    gemm_xwt_kernel<<<rtC, 256, 0, stream>>>(emb_cell, W_cv, Tbuf, NC);
    scatter_kernel<<<(E_cv + 7) / 8, 256, 0, stream>>>(Tbuf, ei_cv, E_cv, agg_v, deg_v);
    // vertex -> cell messages
    gemm_xwt_kernel<<<rtV, 256, 0, stream>>>(emb_vertex, W_vc, Tbuf, NV);
    scatter_kernel<<<(E_vc + 7) / 8, 256, 0, stream>>>(Tbuf, ei_vc, E_vc, agg_c, deg_c);
    // cell -> cell messages
    gemm_xwt_kernel<<<rtC, 256, 0, stream>>>(emb_cell, W_cc, Tbuf, NC);
    scatter_kernel<<<(E_cc + 7) / 8, 256, 0, stream>>>(Tbuf, ei_cc, E_cc, agg_c, deg_c);

    // fused self-linear + mean-message + LayerNorm + GELU
    self_ln_gelu_kernel<<<rtC, 128, 0, stream>>>(emb_cell, W_self_c, b_self_c,
                                                 agg_c, deg_c, g_c, beta_c,
                                                 (float*)d_out, NC);
    self_ln_gelu_kernel<<<rtV, 128, 0, stream>>>(emb_vertex, W_self_v, b_self_v,
                                                 agg_v, deg_v, g_v, beta_v,
                                                 (float*)d_out + nDc, NV);
}